// TIFD_CBAM_38723425140745
// MI455X (gfx1250) — compile-verified
//
#include <hip/hip_runtime.h>
#include <hip/hip_bf16.h>
#include <math.h>

#define B_    8
#define C_    256
#define H_    128
#define W_    128
#define HW_   (H_*W_)
#define CHW_  (C_*HW_)
#define NTOT_ (B_*CHW_)
#define CR_   16
#define EPS_  1e-6f

typedef __attribute__((ext_vector_type(2))) float v2f;
typedef __attribute__((ext_vector_type(8))) float v8f;

__device__ __forceinline__ float nan0f(float v) {
  return (v != v || v > 3.0e38f || v < -3.0e38f) ? 0.0f : v;
}
__device__ __forceinline__ float sigmoidf(float v) { return 1.0f / (1.0f + __expf(-v)); }

__device__ __forceinline__ float waveSum(float v) {
#pragma unroll
  for (int o = 16; o > 0; o >>= 1) v += __shfl_xor(v, o, 32);
  return v;
}
__device__ __forceinline__ float waveMax(float v) {
#pragma unroll
  for (int o = 16; o > 0; o >>= 1) v = fmaxf(v, __shfl_xor(v, o, 32));
  return v;
}
__device__ __forceinline__ float atomicMaxF(float* addr, float val) {
  if (val >= 0.0f)
    return __int_as_float(atomicMax((int*)addr, __float_as_int(val)));
  return __uint_as_float(atomicMin((unsigned int*)addr, __float_as_uint(val)));
}
__device__ __forceinline__ void fillGauss5(float* g) {
  float s = 0.0f;
  for (int i = 0; i < 5; ++i)
    for (int j = 0; j < 5; ++j) {
      float dy = (float)i - 2.0f, dx = (float)j - 2.0f;
      float v = __expf(-(dx * dx + dy * dy) / (2.0f * 1.2f * 1.2f));
      g[i * 5 + j] = v;
      s += v;
    }
  float inv = 1.0f / s;
  for (int k = 0; k < 25; ++k) g[k] *= inv;
}

// ---------------- kernel 0: init per-(b,c) accumulators ----------------
__global__ void k_init(float* sumX, float* maxX, float* sumR, float* sumRdp) {
  int i = blockIdx.x * blockDim.x + threadIdx.x;
  if (i < B_ * C_) {
    sumX[i] = 0.0f;
    sumR[i] = 0.0f;
    sumRdp[i] = 0.0f;
    maxX[i] = -INFINITY;
  }
}

// ---- kernel 1: depthwise gauss 5x5 + |x-gx| + channel/pixel reductions ----
__global__ void k_gauss_stats(const float* __restrict__ x,
                              float* __restrict__ xm, float* __restrict__ rmean,
                              float* __restrict__ sumX, float* __restrict__ maxX,
                              float* __restrict__ sumR) {
  __shared__ float g[25];
  __shared__ float tile[20 * 20];
  __shared__ float xm_acc[256];
  __shared__ float r_acc[256];
  __shared__ float redS[8], redR[8], redM[8];
  const int tid = threadIdx.x;
  if (tid == 0) fillGauss5(g);
  xm_acc[tid] = 0.0f;
  r_acc[tid] = 0.0f;
  const int b = blockIdx.z, ty0 = blockIdx.y * 16, tx0 = blockIdx.x * 16;
  const int ty = tid >> 4, tx = tid & 15;
  const int h = ty0 + ty, w = tx0 + tx;
  __syncthreads();
  for (int ch = 0; ch < C_; ++ch) {
    const float* xp = x + (size_t)(b * C_ + ch) * HW_;
    if (ch + 1 < C_) __builtin_prefetch(xp + HW_ + h * W_ + w, 0, 1);  // gfx1250 global_prefetch_b8
    for (int idx = tid; idx < 400; idx += 256) {
      int r_ = idx / 20, c_ = idx % 20;
      int gh = ty0 + r_ - 2, gw = tx0 + c_ - 2;
      float v = 0.0f;
      if (gh >= 0 && gh < H_ && gw >= 0 && gw < W_) v = xp[gh * W_ + gw];
      tile[idx] = v;
    }
    __syncthreads();
    float acc = 0.0f;
#pragma unroll
    for (int i = 0; i < 5; ++i)
#pragma unroll
      for (int j = 0; j < 5; ++j)
        acc += g[i * 5 + j] * tile[(ty + i) * 20 + (tx + j)];
    float xv = tile[(ty + 2) * 20 + tx + 2];
    float rv = fabsf(xv - acc);
    xm_acc[tid] += xv;
    r_acc[tid] += rv;
    float s1 = waveSum(xv), s2 = waveSum(rv), m1 = waveMax(xv);
    int wid = tid >> 5, lane = tid & 31;
    if (lane == 0) { redS[wid] = s1; redR[wid] = s2; redM[wid] = m1; }
    __syncthreads();
    if (tid == 0) {
      float tS = 0.0f, tR = 0.0f, tM = -INFINITY;
      for (int k = 0; k < 8; ++k) { tS += redS[k]; tR += redR[k]; tM = fmaxf(tM, redM[k]); }
      atomicAdd(&sumX[b * C_ + ch], tS);
      atomicAdd(&sumR[b * C_ + ch], tR);
      atomicMaxF(&maxX[b * C_ + ch], tM);
    }
    __syncthreads();
  }
  xm[(size_t)b * HW_ + h * W_ + w] = xm_acc[tid] * (1.0f / C_);
  rmean[(size_t)b * HW_ + h * W_ + w] = r_acc[tid] * (1.0f / C_);
}

// ---------------- kernel 2a: gm = G1 * xm (pad 2) ----------------
__global__ void k_gm(const float* __restrict__ xm, float* __restrict__ gm) {
  __shared__ float g[25];
  if (threadIdx.x == 0) fillGauss5(g);
  __syncthreads();
  int b = blockIdx.z;
  int h = blockIdx.y * 16 + (threadIdx.x >> 4);
  int w = blockIdx.x * 16 + (threadIdx.x & 15);
  const float* p = xm + (size_t)b * HW_;
  float acc = 0.0f;
  for (int i = 0; i < 5; ++i)
    for (int j = 0; j < 5; ++j) {
      int hh = h + i - 2, ww = w + j - 2;
      if (hh >= 0 && hh < H_ && ww >= 0 && ww < W_) acc += g[i * 5 + j] * p[hh * W_ + ww];
    }
  gm[(size_t)b * HW_ + h * W_ + w] = acc;
}

// ---------------- kernel 2b: usm = USM(3x3) * gm (pad 1) ----------------
__global__ void k_usm(const float* __restrict__ gm, float* __restrict__ usm) {
  int b = blockIdx.z;
  int h = blockIdx.y * 16 + (threadIdx.x >> 4);
  int w = blockIdx.x * 16 + (threadIdx.x & 15);
  const float* p = gm + (size_t)b * HW_;
  auto at = [&](int hh, int ww) -> float {
    return (hh >= 0 && hh < H_ && ww >= 0 && ww < W_) ? p[hh * W_ + ww] : 0.0f;
  };
  float v = 5.0f * at(h, w) - at(h - 1, w) - at(h + 1, w) - at(h, w - 1) - at(h, w + 1);
  usm[(size_t)b * HW_ + h * W_ + w] = v;
}

// ---------------- kernel 3: rdp = |x - usm| stats ----------------
__global__ void k_rdp(const float* __restrict__ x, const float* __restrict__ usm,
                      float* __restrict__ rdpmean, float* __restrict__ sumRdp) {
  __shared__ float red[8];
  int tid = threadIdx.x;
  int b = blockIdx.z;
  int h = blockIdx.y * 16 + (tid >> 4);
  int w = blockIdx.x * 16 + (tid & 15);
  float uv = usm[(size_t)b * HW_ + h * W_ + w];
  float accPix = 0.0f;
  for (int ch = 0; ch < C_; ++ch) {
    float xv = x[(size_t)(b * C_ + ch) * HW_ + h * W_ + w];
    float rdp = fabsf(xv - uv);
    accPix += rdp;
    float s = waveSum(rdp);
    int wid = tid >> 5, lane = tid & 31;
    if (lane == 0) red[wid] = s;
    __syncthreads();
    if (tid == 0) {
      float t = 0.0f;
      for (int k = 0; k < 8; ++k) t += red[k];
      atomicAdd(&sumRdp[b * C_ + ch], t);
    }
    __syncthreads();
  }
  rdpmean[(size_t)b * HW_ + h * W_ + w] = accPix * (1.0f / C_);
}

// ---------------- kernel 4: SE bottleneck via V_WMMA_F32_16X16X4_F32 ----------------
// Rows m = ve*8 + b, ve in {mean(x), max(x), mean(r), mean(rdp)} -> M=32 = two 16-row tiles.
// Layer 1: [32x256] x [256x16]  (64 k-steps/tile).  Layer 2: [32x16] x [16x256] (16 N-tiles).
__global__ void __launch_bounds__(32) k_se(const float* __restrict__ sumX,
                                           const float* __restrict__ maxX,
                                           const float* __restrict__ sumR,
                                           const float* __restrict__ sumRdp,
                                           const float* __restrict__ w1,
                                           const float* __restrict__ w2,
                                           float* __restrict__ ca) {
  __shared__ float h_[32 * 16];
  __shared__ float o2[32 * 256];
  const int lane = threadIdx.x;
  const bool hi = lane >= 16;
  const int lm = lane & 15;
  const float invHW = 1.0f / (float)HW_;
  auto getv = [&](int m, int c) -> float {
    int ve = m >> 3, b = m & 7;
    int idx = b * C_ + c;
    float r;
    if (ve == 0)      r = sumX[idx] * invHW;
    else if (ve == 1) r = maxX[idx];
    else if (ve == 2) r = sumR[idx] * invHW;
    else              r = sumRdp[idx] * invHW;
    return nan0f(r);
  };
  // ---- layer 1 ----
  for (int tile = 0; tile < 2; ++tile) {
    v8f acc = {};
    for (int k0 = 0; k0 < C_; k0 += 4) {
      int m = tile * 16 + lm;
      int ka = hi ? k0 + 2 : k0;
      v2f A; A[0] = getv(m, ka); A[1] = getv(m, ka + 1);
      v2f Bv; Bv[0] = w1[lm * C_ + ka]; Bv[1] = w1[lm * C_ + ka + 1];
      acc = __builtin_amdgcn_wmma_f32_16x16x4_f32(false, A, false, Bv, (short)0, acc,
                                                  false, false);
    }
#pragma unroll
    for (int j = 0; j < 8; ++j) {
      int mloc = j + (hi ? 8 : 0);
      h_[(tile * 16 + mloc) * 16 + lm] = fmaxf(acc[j], 0.0f);  // ReLU
    }
  }
  __syncthreads();
  // ---- layer 2 ----
  for (int tile = 0; tile < 2; ++tile) {
    for (int nt = 0; nt < 16; ++nt) {
      v8f acc = {};
#pragma unroll
      for (int k0 = 0; k0 < 16; k0 += 4) {
        int m = tile * 16 + lm;
        int ka = hi ? k0 + 2 : k0;
        v2f A; A[0] = h_[m * 16 + ka]; A[1] = h_[m * 16 + ka + 1];
        int c = nt * 16 + lm;
        v2f Bv; Bv[0] = w2[c * CR_ + ka]; Bv[1] = w2[c * CR_ + ka + 1];
        acc = __builtin_amdgcn_wmma_f32_16x16x4_f32(false, A, false, Bv, (short)0, acc,
                                                    false, false);
      }
#pragma unroll
      for (int j = 0; j < 8; ++j) {
        int mloc = j + (hi ? 8 : 0);
        o2[(tile * 16 + mloc) * C_ + nt * 16 + lm] = acc[j];
      }
    }
  }
  __syncthreads();
  for (int i = lane; i < B_ * C_; i += 32) {
    int b = i >> 8, c = i & 255;
    float s = o2[b * C_ + c] + o2[(8 + b) * C_ + c] + o2[(16 + b) * C_ + c] +
              o2[(24 + b) * C_ + c];
    ca[i] = sigmoidf(nan0f(s));
  }
}

// ---------------- kernel 5: channel max/avg of x*ca ----------------
__global__ void k_maxavg(const float* __restrict__ x, const float* __restrict__ ca,
                         float* __restrict__ maxc, float* __restrict__ avgc) {
  int tid = threadIdx.x;
  int b = blockIdx.z;
  int h = blockIdx.y * 16 + (tid >> 4);
  int w = blockIdx.x * 16 + (tid & 15);
  float vmax = -INFINITY, vsum = 0.0f;
  for (int ch = 0; ch < C_; ++ch) {
    float v = x[(size_t)(b * C_ + ch) * HW_ + h * W_ + w] * ca[b * C_ + ch];
    vmax = fmaxf(vmax, v);
    vsum += v;
  }
  maxc[(size_t)b * HW_ + h * W_ + w] = vmax;
  avgc[(size_t)b * HW_ + h * W_ + w] = vsum * (1.0f / C_);
}

// ---------------- kernel 6a: coherence + phase-congruency proxy ----------------
__global__ void k_cohpc(const float* __restrict__ avgc, float* __restrict__ coh,
                        float* __restrict__ pc) {
  __shared__ float t[20 * 20];
  __shared__ float gxs[18 * 18];
  __shared__ float gys[18 * 18];
  int tid = threadIdx.x;
  int b = blockIdx.z, ty0 = blockIdx.y * 16, tx0 = blockIdx.x * 16;
  const float* p = avgc + (size_t)b * HW_;
  for (int idx = tid; idx < 400; idx += 256) {
    int r_ = idx / 20, c_ = idx % 20;
    int gh = ty0 + r_ - 2, gw = tx0 + c_ - 2;
    t[idx] = (gh >= 0 && gh < H_ && gw >= 0 && gw < W_) ? nan0f(p[gh * W_ + gw]) : 0.0f;
  }
  __syncthreads();
  for (int idx = tid; idx < 324; idx += 256) {
    int r_ = idx / 18, c_ = idx % 18;  // pixel at tile coord (r_+1, c_+1)
    int gh = ty0 + r_ - 1, gw = tx0 + c_ - 1;
    float gx = 0.0f, gy = 0.0f;
    if (gh >= 0 && gh < H_ && gw >= 0 && gw < W_) {
      float a00 = t[r_ * 20 + c_], a01 = t[r_ * 20 + c_ + 1], a02 = t[r_ * 20 + c_ + 2];
      float a10 = t[(r_ + 1) * 20 + c_], a12 = t[(r_ + 1) * 20 + c_ + 2];
      float a20 = t[(r_ + 2) * 20 + c_], a21 = t[(r_ + 2) * 20 + c_ + 1],
            a22 = t[(r_ + 2) * 20 + c_ + 2];
      gx = -a00 + a02 - 2.0f * a10 + 2.0f * a12 - a20 + a22;
      gy = -a00 - 2.0f * a01 - a02 + a20 + 2.0f * a21 + a22;
    }
    gxs[idx] = gx;
    gys[idx] = gy;
  }
  __syncthreads();
  int ty = tid >> 4, tx = tid & 15;
  int h = ty0 + ty, w = tx0 + tx;
  float J11 = 0.0f, J22 = 0.0f, J12 = 0.0f;
#pragma unroll
  for (int dy = -1; dy <= 1; ++dy)
#pragma unroll
    for (int dx = -1; dx <= 1; ++dx) {
      float gx = gxs[(ty + 1 + dy) * 18 + tx + 1 + dx];
      float gy = gys[(ty + 1 + dy) * 18 + tx + 1 + dx];
      J11 += gx * gx;
      J22 += gy * gy;
      J12 += gx * gy;
    }
  J11 *= (1.0f / 9.0f); J22 *= (1.0f / 9.0f); J12 *= (1.0f / 9.0f);
  float trace = J11 + J22;
  float det = J11 * J22 - J12 * J12;
  float tmp = sqrtf(fmaxf(trace * trace - 4.0f * det, 0.0f));
  float cohv = tmp / (trace + 1e-6f);
  float gxc = gxs[(ty + 1) * 18 + tx + 1], gyc = gys[(ty + 1) * 18 + tx + 1];
  float gmag = sqrtf(gxc * gxc + gyc * gyc + 1e-12f);
  float c00 = t[(ty + 2) * 20 + tx + 2];
  float lap = fabsf(t[(ty + 1) * 20 + tx + 2] + t[(ty + 3) * 20 + tx + 2] +
                    t[(ty + 2) * 20 + tx + 1] + t[(ty + 2) * 20 + tx + 3] - 4.0f * c00);
  float pcv = gmag / (gmag + lap + 1e-6f);
  coh[(size_t)b * HW_ + h * W_ + w] = cohv;
  pc[(size_t)b * HW_ + h * W_ + w] = pcv;
}

// ---------------- kernel 6b: per-(map,b) mean / unbiased std ----------------
__global__ void k_normstats(const float* __restrict__ maxc, const float* __restrict__ avgc,
                            const float* __restrict__ rmean, const float* __restrict__ coh,
                            const float* __restrict__ pc, const float* __restrict__ rdpmean,
                            float* __restrict__ meanS, float* __restrict__ stdS) {
  __shared__ float redA[8], redB[8];
  int b = blockIdx.x, map = blockIdx.y;
  const float* p;
  switch (map) {
    case 0: p = maxc; break;
    case 1: p = avgc; break;
    case 2: p = rmean; break;
    case 3: p = coh; break;
    case 4: p = pc; break;
    default: p = rdpmean; break;
  }
  p += (size_t)b * HW_;
  int tid = threadIdx.x;
  float s = 0.0f, s2 = 0.0f;
  for (int i = tid; i < HW_; i += 256) {
    float v = p[i];
    s += v;
    s2 += v * v;
  }
  s = waveSum(s);
  s2 = waveSum(s2);
  int wid = tid >> 5, lane = tid & 31;
  if (lane == 0) { redA[wid] = s; redB[wid] = s2; }
  __syncthreads();
  if (tid == 0) {
    float tS = 0.0f, t2 = 0.0f;
    for (int k = 0; k < 8; ++k) { tS += redA[k]; t2 += redB[k]; }
    float mean = tS / (float)HW_;
    float var = (t2 - (float)HW_ * mean * mean) / (float)(HW_ - 1);
    float sd = sqrtf(fmaxf(var, 0.0f)) + EPS_;
    meanS[map * B_ + b] = mean;
    stdS[map * B_ + b] = sd;
  }
}

// ---------------- kernel 6c: 7x7 conv over 6 normalized maps -> sigmoid ----------------
__global__ void k_sa(const float* __restrict__ maxc, const float* __restrict__ avgc,
                     const float* __restrict__ rmean, const float* __restrict__ coh,
                     const float* __restrict__ pc, const float* __restrict__ rdpmean,
                     const float* __restrict__ meanS, const float* __restrict__ stdS,
                     const float* __restrict__ wsp, float* __restrict__ sa) {
  __shared__ float tiles[6][22 * 22];
  __shared__ float wk[6 * 49];
  int tid = threadIdx.x;
  int b = blockIdx.z, ty0 = blockIdx.y * 16, tx0 = blockIdx.x * 16;
  for (int i = tid; i < 294; i += 256) wk[i] = wsp[i];
  const float* maps[6] = {maxc, avgc, rmean, coh, pc, rdpmean};
#pragma unroll
  for (int m = 0; m < 6; ++m) {
    const float* p = maps[m] + (size_t)b * HW_;
    float mean = meanS[m * B_ + b];
    float inv = 1.0f / stdS[m * B_ + b];
    for (int idx = tid; idx < 484; idx += 256) {
      int r_ = idx / 22, c_ = idx % 22;
      int gh = ty0 + r_ - 3, gw = tx0 + c_ - 3;
      float v = 0.0f;
      if (gh >= 0 && gh < H_ && gw >= 0 && gw < W_) v = nan0f((p[gh * W_ + gw] - mean) * inv);
      tiles[m][idx] = v;
    }
  }
  __syncthreads();
  int ty = tid >> 4, tx = tid & 15;
  float acc = 0.0f;
#pragma unroll
  for (int m = 0; m < 6; ++m)
    for (int ky = 0; ky < 7; ++ky)
#pragma unroll
      for (int kx = 0; kx < 7; ++kx)
        acc += wk[(m * 7 + ky) * 7 + kx] * tiles[m][(ty + ky) * 22 + tx + kx];
  sa[(size_t)b * HW_ + (ty0 + ty) * W_ + tx0 + tx] = nan0f(sigmoidf(acc));
}

// ---------------- kernel 7: out = x + nan0(x*ca*sa), float4 streamed ----------------
__global__ void k_final(const float* __restrict__ x, const float* __restrict__ ca,
                        const float* __restrict__ sa, float* __restrict__ out) {
  size_t i4 = (size_t)blockIdx.x * blockDim.x + threadIdx.x;
  size_t e = i4 * 4;
  if (e >= (size_t)NTOT_) return;
  const float4 xv = ((const float4*)x)[i4];
  int b = (int)(e / CHW_);
  int c = (int)((e / HW_) % C_);
  int hw = (int)(e % HW_);
  float cav = ca[b * C_ + c];
  const float4 sav = ((const float4*)(sa + (size_t)b * HW_))[hw >> 2];
  float4 o;
  o.x = xv.x + nan0f(xv.x * cav * sav.x);
  o.y = xv.y + nan0f(xv.y * cav * sav.y);
  o.z = xv.z + nan0f(xv.z * cav * sav.z);
  o.w = xv.w + nan0f(xv.w * cav * sav.w);
  ((float4*)out)[i4] = o;
}

extern "C" void kernel_launch(void* const* d_in, const int* in_sizes, int n_in,
                              void* d_out, int out_size, void* d_ws, size_t ws_size,
                              hipStream_t stream) {
  const float* x = (const float*)d_in[0];
  const float* w1 = (const float*)d_in[1];   // (16, 256)
  const float* w2 = (const float*)d_in[2];   // (256, 16)
  const float* wsp = (const float*)d_in[3];  // (1, 6, 7, 7)
  float* out = (float*)d_out;

  float* ws = (float*)d_ws;
  float* xm = ws;      ws += B_ * HW_;
  float* rmean = ws;   ws += B_ * HW_;
  float* gm = ws;      ws += B_ * HW_;
  float* usm = ws;     ws += B_ * HW_;
  float* rdpmean = ws; ws += B_ * HW_;
  float* maxc = ws;    ws += B_ * HW_;
  float* avgc = ws;    ws += B_ * HW_;
  float* coh = ws;     ws += B_ * HW_;
  float* pc = ws;      ws += B_ * HW_;
  float* sa = ws;      ws += B_ * HW_;
  float* sumX = ws;    ws += B_ * C_;
  float* maxX = ws;    ws += B_ * C_;
  float* sumR = ws;    ws += B_ * C_;
  float* sumRdp = ws;  ws += B_ * C_;
  float* ca = ws;      ws += B_ * C_;
  float* meanS = ws;   ws += 6 * B_;
  float* stdS = ws;    ws += 6 * B_;

  dim3 tiles(W_ / 16, H_ / 16, B_);
  k_init<<<(B_ * C_ + 255) / 256, 256, 0, stream>>>(sumX, maxX, sumR, sumRdp);
  k_gauss_stats<<<tiles, 256, 0, stream>>>(x, xm, rmean, sumX, maxX, sumR);
  k_gm<<<tiles, 256, 0, stream>>>(xm, gm);
  k_usm<<<tiles, 256, 0, stream>>>(gm, usm);
  k_rdp<<<tiles, 256, 0, stream>>>(x, usm, rdpmean, sumRdp);
  k_se<<<1, 32, 0, stream>>>(sumX, maxX, sumR, sumRdp, w1, w2, ca);
  k_maxavg<<<tiles, 256, 0, stream>>>(x, ca, maxc, avgc);
  k_cohpc<<<tiles, 256, 0, stream>>>(avgc, coh, pc);
  k_normstats<<<dim3(B_, 6, 1), 256, 0, stream>>>(maxc, avgc, rmean, coh, pc, rdpmean,
                                                  meanS, stdS);
  k_sa<<<tiles, 256, 0, stream>>>(maxc, avgc, rmean, coh, pc, rdpmean, meanS, stdS, wsp, sa);
  k_final<<<NTOT_ / 4 / 256, 256, 0, stream>>>(x, ca, sa, out);
}